// DiscreteVariableQNN_55516747268623
// MI455X (gfx1250) — compile-verified
//
#include <hip/hip_runtime.h>
#include <hip/hip_bf16.h>

// ---------------------------------------------------------------------------
// 16-qubit state-vector QNN on gfx1250 (MI455X), fp32 WMMA formulation.
//
// Math: per layer, the 48 RX rotations collapse (RX(a)RX(b)RX(c)=RX(a+b+c))
// into Kron of 16 RX gates == 4 complex 16x16 matrices G0..G3 applied along
// the 4 hex digits of the amplitude index; the CNOT chain is the prefix-XOR
// bit permutation (inverse-Gray scatter). All matmuls via v_wmma_f32_16x16x4.
// ---------------------------------------------------------------------------

typedef float v2f __attribute__((ext_vector_type(2)));
typedef float v8f __attribute__((ext_vector_type(8)));

#define NQ 16
#define NSTATE 65536           // 2^16 amplitudes
#define BATCH 512
#define NLAYERS 4

// ---- WMMA helpers ----------------------------------------------------------
__device__ __forceinline__ v8f wmma4(v2f a, v2f b, v8f c) {
  // D = A(16x4) * B(4x16) + C ; fp32. neg_a/neg_b must be 0 for F32 (ISA 7.12).
  return __builtin_amdgcn_wmma_f32_16x16x4_f32(false, a, false, b, (short)0, c,
                                               false, false);
}

// A-fragment (row m) of a row-major 16x16 matrix, K-chunk k.
// Identical formula also yields the B-fragment of M^T (B[kk][n] = M[n][kk]).
__device__ __forceinline__ v2f frag_rm(const float* M, int m, int hs, int k,
                                       bool neg) {
  int c0 = 4 * k + (hs ? 2 : 0);
  int c1 = 4 * k + (hs ? 3 : 1);
  v2f r;
  r.x = M[m * 16 + c0];
  r.y = M[m * 16 + c1];
  if (neg) { r.x = -r.x; r.y = -r.y; }
  return r;
}

// B-fragment of a row-major K x N (16x16) matrix: B[kk][n], n = m.
__device__ __forceinline__ v2f frag_kn(const float* M, int m, int hs, int k,
                                       bool neg) {
  int r0 = 4 * k + (hs ? 2 : 0);
  int r1 = 4 * k + (hs ? 3 : 1);
  v2f r;
  r.x = M[r0 * 16 + m];
  r.y = M[r1 * 16 + m];
  if (neg) { r.x = -r.x; r.y = -r.y; }
  return r;
}

// ---- Kernel 0: build the 16 per-(layer,group) complex 16x16 gate matrices --
// gate entry(i,j) = prod_m [ bits match ? cos(h_q) : sin(h_q) ] * (-i)^{#flips}
__global__ __launch_bounds__(256) void qnn_gates(const float* __restrict__ params,
                                                 float* __restrict__ gR,
                                                 float* __restrict__ gI) {
  int L = blockIdx.x >> 2;
  int g = blockIdx.x & 3;
  int i = threadIdx.x >> 4;
  int j = threadIdx.x & 15;
  const float* p = params + L * (NQ * 3);
  float mag = 1.0f;
  int flips = 0;
  #pragma unroll
  for (int m = 0; m < 4; ++m) {
    int q = 4 * g + m;
    float h = 0.5f * (p[3 * q] + p[3 * q + 1] + p[3 * q + 2]);
    int bit = 3 - m;  // qubit 4g+m is the MSB of the digit
    int ib = (i >> bit) & 1, jb = (j >> bit) & 1;
    if (ib == jb) mag *= __cosf(h) * 0.0f + cosf(h);
    else { mag *= sinf(h); ++flips; }
  }
  float re = 0.f, im = 0.f;
  switch (flips & 3) {           // (-i)^flips
    case 0: re =  mag; break;
    case 1: im = -mag; break;
    case 2: re = -mag; break;
    case 3: im =  mag; break;
  }
  int o = blockIdx.x * 256 + i * 16 + j;
  gR[o] = re;
  gI[o] = im;
}

// ---- Kernel 1: encode: amp(idx) = (1/256)*(-1)^popc(idx & mask), imag = 0 --
__global__ __launch_bounds__(256) void qnn_encode(const float* __restrict__ x,
                                                  float* __restrict__ sR,
                                                  float* __restrict__ sI) {
  __shared__ unsigned mask;
  int b = blockIdx.y;
  if (threadIdx.x == 0) {
    unsigned m = 0;
    #pragma unroll
    for (int q = 0; q < NQ; ++q)
      if (x[b * NQ + q] > 0.0f) m |= 1u << (15 - q);
    mask = m;
  }
  __syncthreads();
  unsigned idx = blockIdx.x * 256 + threadIdx.x;
  float amp = (__popc(idx & mask) & 1) ? -0.00390625f : 0.00390625f;
  size_t o = (size_t)b * NSTATE + idx;
  sR[o] = amp;
  sI[o] = 0.0f;
}

// ---- Kernel 2: apply a pair of group gates (complex 16x16) per wave --------
// high==0: tiles are 256 contiguous amps (digits 2,3); write back in place.
// high==1: tiles are stride-256 views (digits 0,1); scatter through the
//          CNOT-chain permutation y = inverseGray(x).
__global__ __launch_bounds__(256) void qnn_apply(const float* __restrict__ inR,
                                                 const float* __restrict__ inI,
                                                 float* __restrict__ outR,
                                                 float* __restrict__ outI,
                                                 const float* __restrict__ gAr,
                                                 const float* __restrict__ gAi,
                                                 const float* __restrict__ gBr,
                                                 const float* __restrict__ gBi,
                                                 int high) {
  __shared__ __align__(16) float ls[8][2][256];  // per-wave S tile (re, im)
  __shared__ __align__(16) float lt[8][2][256];  // per-wave T staging
  const int lane = threadIdx.x & 31;
  const int wave = threadIdx.x >> 5;
  const int task = blockIdx.x * 8 + wave;   // 512*256 tasks exactly
  const int b = task >> 8;
  const int blk = task & 255;               // low: tile id ; high: lo value
  const size_t base = (size_t)b * NSTATE;

  float* S_r = ls[wave][0];
  float* S_i = ls[wave][1];
  float* T_r = lt[wave][0];
  float* T_i = lt[wave][1];

  // ---- load one 16x16 complex tile into LDS ----
  if (!high) {
    const float4* pr = (const float4*)(inR + base + (size_t)blk * 256);
    const float4* pi = (const float4*)(inI + base + (size_t)blk * 256);
    ((float4*)S_r)[lane]      = pr[lane];
    ((float4*)S_r)[lane + 32] = pr[lane + 32];
    ((float4*)S_i)[lane]      = pi[lane];
    ((float4*)S_i)[lane + 32] = pi[lane + 32];
  } else {
    #pragma unroll
    for (int t = 0; t < 8; ++t) {
      int e = lane + 32 * t;  // e = d0*16+d1 over index bits 15..8
      S_r[e] = inR[base + (size_t)e * 256 + blk];
      S_i[e] = inI[base + (size_t)e * 256 + blk];
    }
  }
  __syncthreads();

  const int m = lane & 15;
  const int hs = lane >> 4;

  // ---- T = S * GB^T  (complex; right gate acts on the column digit) ----
  v8f tr = {}; v8f ti = {};
  #pragma unroll
  for (int k = 0; k < 4; ++k)
    tr = wmma4(frag_rm(S_r, m, hs, k, false), frag_rm(gBr, m, hs, k, false), tr);
  #pragma unroll
  for (int k = 0; k < 4; ++k)   // - Si*Gi : negate the gate fragment
    tr = wmma4(frag_rm(S_i, m, hs, k, false), frag_rm(gBi, m, hs, k, true), tr);
  #pragma unroll
  for (int k = 0; k < 4; ++k)
    ti = wmma4(frag_rm(S_r, m, hs, k, false), frag_rm(gBi, m, hs, k, false), ti);
  #pragma unroll
  for (int k = 0; k < 4; ++k)
    ti = wmma4(frag_rm(S_i, m, hs, k, false), frag_rm(gBr, m, hs, k, false), ti);

  // ---- restage T: C/D layout -> row-major in LDS ----
  #pragma unroll
  for (int v = 0; v < 8; ++v) {
    int r = v + 8 * hs;
    T_r[r * 16 + m] = tr[v];
    T_i[r * 16 + m] = ti[v];
  }
  __syncthreads();

  // ---- D = GA * T  (complex; left gate acts on the row digit) ----
  v8f dr = {}; v8f di = {};
  #pragma unroll
  for (int k = 0; k < 4; ++k)
    dr = wmma4(frag_rm(gAr, m, hs, k, false), frag_kn(T_r, m, hs, k, false), dr);
  #pragma unroll
  for (int k = 0; k < 4; ++k)   // - Gi*Ti : negate the gate fragment
    dr = wmma4(frag_rm(gAi, m, hs, k, true), frag_kn(T_i, m, hs, k, false), dr);
  #pragma unroll
  for (int k = 0; k < 4; ++k)
    di = wmma4(frag_rm(gAr, m, hs, k, false), frag_kn(T_i, m, hs, k, false), di);
  #pragma unroll
  for (int k = 0; k < 4; ++k)
    di = wmma4(frag_rm(gAi, m, hs, k, false), frag_kn(T_r, m, hs, k, false), di);

  // ---- write back ----
  if (!high) {
    __syncthreads();  // done with S as input; reuse as output staging
    #pragma unroll
    for (int v = 0; v < 8; ++v) {
      int r = v + 8 * hs;
      S_r[r * 16 + m] = dr[v];
      S_i[r * 16 + m] = di[v];
    }
    __syncthreads();
    float4* qr = (float4*)(outR + base + (size_t)blk * 256);
    float4* qi = (float4*)(outI + base + (size_t)blk * 256);
    qr[lane]      = ((float4*)S_r)[lane];
    qr[lane + 32] = ((float4*)S_r)[lane + 32];
    qi[lane]      = ((float4*)S_i)[lane];
    qi[lane + 32] = ((float4*)S_i)[lane + 32];
  } else {
    // scatter through the CNOT-chain permutation: y_b = XOR of x bits >= b
    #pragma unroll
    for (int v = 0; v < 8; ++v) {
      int r = v + 8 * hs;
      unsigned xidx = (unsigned)((r * 16 + m) * 256 + blk);
      unsigned y = xidx;
      y ^= y >> 1; y ^= y >> 2; y ^= y >> 4; y ^= y >> 8;
      outR[base + y] = dr[v];
      outI[base + y] = di[v];
    }
  }
}

// ---- Kernel 3: score[b] = 1 - sum |amp|^2 ---------------------------------
__global__ __launch_bounds__(256) void qnn_reduce(const float* __restrict__ sR,
                                                  const float* __restrict__ sI,
                                                  float* __restrict__ out) {
  __shared__ float red[256];
  int b = blockIdx.x;
  size_t base = (size_t)b * NSTATE;
  float s = 0.0f;
  for (int i = threadIdx.x; i < NSTATE; i += 256) {
    float r = sR[base + i], im = sI[base + i];
    s += r * r + im * im;
  }
  red[threadIdx.x] = s;
  __syncthreads();
  for (int off = 128; off > 0; off >>= 1) {
    if (threadIdx.x < off) red[threadIdx.x] += red[threadIdx.x + off];
    __syncthreads();
  }
  if (threadIdx.x == 0) out[b] = 1.0f - red[0];
}

// ---------------------------------------------------------------------------
extern "C" void kernel_launch(void* const* d_in, const int* in_sizes, int n_in,
                              void* d_out, int out_size, void* d_ws, size_t ws_size,
                              hipStream_t stream) {
  const float* x = (const float*)d_in[0];       // [512, 16]
  const float* params = (const float*)d_in[1];  // [4, 48]
  float* out = (float*)d_out;                   // [512]

  const size_t NST = (size_t)BATCH * NSTATE;    // floats per plane
  char* w = (char*)d_ws;
  float* gR = (float*)w;                        // 16 * 256 floats
  float* gI = gR + 16 * 256;
  float* AR = (float*)(w + (1 << 15));          // ping state, real plane
  float* AI = AR + NST;
  float* BR = AI + NST;                         // pong state
  float* BI = BR + NST;

  qnn_gates<<<16, 256, 0, stream>>>(params, gR, gI);
  qnn_encode<<<dim3(256, BATCH), 256, 0, stream>>>(x, AR, AI);

  for (int L = 0; L < NLAYERS; ++L) {
    const float* g0r = gR + (L * 4 + 0) * 256; const float* g0i = gI + (L * 4 + 0) * 256;
    const float* g1r = gR + (L * 4 + 1) * 256; const float* g1i = gI + (L * 4 + 1) * 256;
    const float* g2r = gR + (L * 4 + 2) * 256; const float* g2i = gI + (L * 4 + 2) * 256;
    const float* g3r = gR + (L * 4 + 3) * 256; const float* g3i = gI + (L * 4 + 3) * 256;
    // digits 2,3 (low 8 index bits), tile-local
    qnn_apply<<<(BATCH * 256) / 8, 256, 0, stream>>>(AR, AI, BR, BI,
                                                     g2r, g2i, g3r, g3i, 0);
    // digits 0,1 (high 8 index bits) + CNOT-chain permutation scatter
    qnn_apply<<<(BATCH * 256) / 8, 256, 0, stream>>>(BR, BI, AR, AI,
                                                     g0r, g0i, g1r, g1i, 1);
  }

  qnn_reduce<<<BATCH, 256, 0, stream>>>(AR, AI, out);
}